// WindowAttention3d3_65524021068051
// MI455X (gfx1250) — compile-verified
//
#include <hip/hip_runtime.h>
#include <hip/hip_bf16.h>

typedef __attribute__((ext_vector_type(2))) float v2f;
typedef __attribute__((ext_vector_type(8))) float v8f;

#define WMMA_F32(a, b, c) \
    __builtin_amdgcn_wmma_f32_16x16x4_f32(false, (a), false, (b), (short)0, (c), false, false)

// ---- CDNA5 async global->LDS copy (ASYNCcnt-tracked), with graceful fallback ----
#if defined(__has_builtin)
#if __has_builtin(__builtin_amdgcn_global_load_async_to_lds_b128)
#define HAVE_ASYNC_LDS 1
#endif
#if __has_builtin(__builtin_amdgcn_tensor_load_to_lds)
#define HAVE_TDM 1
#endif
#endif
#ifndef HAVE_ASYNC_LDS
#define HAVE_ASYNC_LDS 0
#endif
#ifndef HAVE_TDM
#define HAVE_TDM 0
#endif

#if HAVE_ASYNC_LDS
// Builtin signature (per hipcc diagnostic): pointers to 16-byte int vectors.
typedef int a16_t __attribute__((vector_size(16)));
__device__ __forceinline__ void async_copy16(const float* gsrc, float* ldst) {
    // GLOBAL_LOAD_ASYNC_TO_LDS_B128: per-lane 16B, global -> LDS, no VGPR staging
    __builtin_amdgcn_global_load_async_to_lds_b128((a16_t*)gsrc, (a16_t*)ldst, 0, 0);
}
__device__ __forceinline__ void async_wait0() {
#if __has_builtin(__builtin_amdgcn_s_wait_asynccnt)
    __builtin_amdgcn_s_wait_asynccnt(0);
#else
    asm volatile("s_wait_asynccnt 0x0" ::: "memory");
#endif
}
#else
__device__ __forceinline__ void async_wait0() {}
#endif

#if HAVE_TDM
// ---- Tensor Data Mover: 2D strided tile global->LDS with per-row LDS padding.
// Descriptor bit layout per CDNA5 ISA ch.8 (D# groups).  This toolchain has
// the 6-arg builtin: (u32x4 g0, i32x8 g1, i32x4 g2, i32x4 g3, i32x8, i32 cpol).
typedef unsigned int u32x4 __attribute__((vector_size(16)));
typedef int i32x8 __attribute__((vector_size(32)));
typedef int i32x4 __attribute__((vector_size(16)));

__device__ __forceinline__ void tdm_load_2d_f32(
    const float* gsrc, const float* lds_dst,
    unsigned tensor_w, unsigned tensor_h,       // tensor dims (elements)
    unsigned tile_w, unsigned tile_h,           // tile dims (elements)
    unsigned row_stride_elems,                  // tensor_dim0_stride
    unsigned pad_interval_code,                 // 4 -> pad after every 32 dwords
    unsigned pad_amount_code)                   // 3 -> 4 dwords ; 7 -> 8 dwords
{
    const unsigned long long ga = (unsigned long long)(uintptr_t)gsrc;
    const unsigned lds_off = (unsigned)(uintptr_t)lds_dst;   // low 32 = LDS byte offset

    u32x4 g0;
    g0[0] = 1u;                                   // count=1 valid descriptor
    g0[1] = lds_off;                              // lds_addr (bytes)
    g0[2] = (unsigned)(ga & 0xFFFFFFFFu);         // global_addr[31:0]
    g0[3] = (unsigned)((ga >> 32) & 0x1FFFFFFu)   // global_addr[56:32]
          | (2u << 30);                           // D#.type = 2 ("image")

    i32x8 g1;
    g1[0] = (int)((2u << 16)                      // data_size = 4B
          | (1u << 20)                            // pad_enable
          | (pad_interval_code << 22)
          | (pad_amount_code << 25));
    g1[1] = (int)((tensor_w & 0xFFFFu) << 16);    // tensor_dim0[15:0] -> bits 63:48
    g1[2] = (int)(((tensor_w >> 16) & 0xFFFFu)    // tensor_dim0[31:16]
          | ((tensor_h & 0xFFFFu) << 16));        // tensor_dim1[15:0]
    g1[3] = (int)(((tensor_h >> 16) & 0xFFFFu)    // tensor_dim1[31:16]
          | ((tile_w & 0xFFFFu) << 16));          // tile_dim0
    g1[4] = (int)(tile_h & 0xFFFFu);              // tile_dim1 ; tile_dim2 = 0
    g1[5] = (int)row_stride_elems;                // tensor_dim0_stride[31:0]
    g1[6] = 0;                                    // stride hi / dim1_stride
    g1[7] = 0;

    const i32x4 gz4 = {0, 0, 0, 0};               // 2D tensor: groups 2/3 unused
    const i32x8 gz8 = {0, 0, 0, 0, 0, 0, 0, 0};
    __builtin_amdgcn_tensor_load_to_lds(g0, g1, gz4, gz4, gz8, 0);
}
__device__ __forceinline__ void tensor_wait0() {
#if __has_builtin(__builtin_amdgcn_s_wait_tensorcnt)
    __builtin_amdgcn_s_wait_tensorcnt(0);
#else
    asm volatile("s_wait_tensorcnt 0x0" ::: "memory");
#endif
}
#endif

// ---------------------------------------------------------------------------
// Generic C = A(MxK) @ B(KxN) + bias(N), fp32, row-major, using
// V_WMMA_F32_16X16X4_F32.  Requires K % 32 == 0, N % 128 == 0.
// Block: 256 threads (8 waves), 64x128 tile; each wave does 32x32.
// Double-buffered LDS staging via async global->LDS copies: the stage for
// k0+32 is issued before computing k0, so HBM latency hides under WMMA.
// ---------------------------------------------------------------------------
__global__ __launch_bounds__(256) void gemm_wmma_f32(
    const float* __restrict__ A, const float* __restrict__ B,
    const float* __restrict__ bias, float* __restrict__ C,
    int M, int N, int K)
{
    __shared__ float As[2][64 * 36];    // row pitch 144B: 16B aligned, conflict-free col reads
    __shared__ float Bs[2][32 * 136];   // row pitch 544B: lane-half row pairs hit disjoint banks

    const int tid  = threadIdx.x;
    const int lane = tid & 31;
    const int wave = tid >> 5;            // 0..7
    const int wm   = (wave >> 2) * 32;    // 0 or 32
    const int wn   = (wave & 3) * 32;     // 0,32,64,96
    const int half = lane >> 4;           // lane-half selects K pair
    const int l16  = lane & 15;

    const int m_blk = blockIdx.y * 64;
    const int n_blk = blockIdx.x * 128;

    v8f acc[2][2] = {};

    auto stage = [&](int buf, int k0) {
        float* Asb = &As[buf][0];
        float* Bsb = &Bs[buf][0];
#if HAVE_ASYNC_LDS
        if (m_blk + 64 <= M) {
            for (int i = tid; i < 64 * 8; i += 256) {
                const int r = i >> 3, c4 = (i & 7) * 4;
                async_copy16(A + (size_t)(m_blk + r) * K + k0 + c4, &Asb[r * 36 + c4]);
            }
        } else
#endif
        {
            for (int i = tid; i < 64 * 8; i += 256) {
                const int r = i >> 3, c4 = (i & 7) * 4;
                const int gr = m_blk + r;
                float4 v = make_float4(0.f, 0.f, 0.f, 0.f);
                if (gr < M) v = *(const float4*)(A + (size_t)gr * K + k0 + c4);
                *(float4*)&Asb[r * 36 + c4] = v;
            }
        }
#if HAVE_ASYNC_LDS
        for (int i = tid; i < 32 * 32; i += 256) {
            const int r = i >> 5, c4 = (i & 31) * 4;
            async_copy16(B + (size_t)(k0 + r) * N + n_blk + c4, &Bsb[r * 136 + c4]);
        }
#else
        for (int i = tid; i < 32 * 32; i += 256) {
            const int r = i >> 5, c4 = (i & 31) * 4;
            const float4 v = *(const float4*)(B + (size_t)(k0 + r) * N + n_blk + c4);
            *(float4*)&Bsb[r * 136 + c4] = v;
        }
#endif
    };

    stage(0, 0);
    int buf = 0;
    for (int k0 = 0; k0 < K; k0 += 32, buf ^= 1) {
        async_wait0();              // my async copies (issued last iter) landed
        __syncthreads();            // everyone's copies landed; prev compute done
        if (k0 + 32 < K) stage(buf ^ 1, k0 + 32);   // prefetch next tile async

        const float* Asb = &As[buf][0];
        const float* Bsb = &Bs[buf][0];
        #pragma unroll
        for (int kk = 0; kk < 32; kk += 4) {
            const int ko = kk + 2 * half;       // K pair owned by this lane half
            v2f af[2], bf[2];
            #pragma unroll
            for (int mt = 0; mt < 2; ++mt) {
                const float* ap = &Asb[(wm + mt * 16 + l16) * 36 + ko];
                af[mt] = (v2f){ap[0], ap[1]};
            }
            #pragma unroll
            for (int nt = 0; nt < 2; ++nt) {
                const int nc = wn + nt * 16 + l16;
                bf[nt] = (v2f){Bsb[ko * 136 + nc], Bsb[(ko + 1) * 136 + nc]};
            }
            #pragma unroll
            for (int mt = 0; mt < 2; ++mt)
                #pragma unroll
                for (int nt = 0; nt < 2; ++nt)
                    acc[mt][nt] = WMMA_F32(af[mt], bf[nt], acc[mt][nt]);
        }
        __syncthreads();            // done reading buf before it is restaged
    }

    // ---- store + bias (C VGPR j -> rows j / j+8 per lane half) ----
    #pragma unroll
    for (int mt = 0; mt < 2; ++mt) {
        #pragma unroll
        for (int nt = 0; nt < 2; ++nt) {
            const int col = n_blk + wn + nt * 16 + l16;
            const float bv = bias[col];
            #pragma unroll
            for (int j = 0; j < 8; ++j) {
                const int row = m_blk + wm + mt * 16 + j + 8 * half;
                if (row < M)
                    C[(size_t)row * N + col] = acc[mt][nt][j] + bv;
            }
        }
    }
}

// ---------------------------------------------------------------------------
// Fused per-(window, head) attention.  blockIdx.x = b*16 + h.
// 128 threads = 4 waves; wave w owns query rows [16w, 16w+16).
// S = q @ k^T ; S = S*scale + relpos_bias ; softmax ; O = P @ v ; scatter.
// q/k/v tiles staged by the Tensor Data Mover (one TENSOR_LOAD_TO_LDS each,
// hardware row padding reproduces the banked LDS pitch); scale folded into
// the S-writeback.
// ---------------------------------------------------------------------------
__global__ __launch_bounds__(128) void attn_wmma_f32(
    const float* __restrict__ qkv,     // (nWB*49, 1536) = [q|k|v] per head
    const float* __restrict__ table,   // (169, 16) rel-pos bias table
    float* __restrict__ obuf)          // (nWB*49, 512)
{
    __shared__ float qs[64 * 36];
    __shared__ float ks[64 * 36];
    __shared__ float vs[64 * 40];
    __shared__ float ps[64 * 68];

    const int b = blockIdx.x >> 4;
    const int h = blockIdx.x & 15;
    const int tid  = threadIdx.x;
    const int lane = tid & 31;
    const int wave = tid >> 5;
    const int half = lane >> 4;
    const int l16  = lane & 15;
    const int m0   = wave * 16;
    const float scale = 0.17677669529663687f;  // 1/sqrt(32)

    const float* base = qkv + (size_t)b * 49 * 1536 + h * 32;

#if HAVE_TDM
    // ---- TDM: 49x32 f32 tile, row stride 1536 elems; LDS pitch via pad:
    // pad_interval=4 (32 dwords), pad_amount=3 (4 dw) -> 36-float pitch,
    // pad_amount=7 (8 dw) -> 40-float pitch.  Wave 0 issues; TDM ignores EXEC.
    if (wave == 0) {
        tdm_load_2d_f32(base,         qs, 32, 49, 32, 49, 1536, 4, 3);
        tdm_load_2d_f32(base + 512,   ks, 32, 49, 32, 49, 1536, 4, 3);
        tdm_load_2d_f32(base + 1024,  vs, 32, 49, 32, 49, 1536, 4, 7);
    }
    for (int i = tid; i < 15 * 8; i += 128) {   // zero-pad rows 49..63
        const int r = 49 + (i >> 3), c4 = (i & 7) * 4;
        const float4 z = make_float4(0.f, 0.f, 0.f, 0.f);
        *(float4*)&qs[r * 36 + c4] = z;
        *(float4*)&ks[r * 36 + c4] = z;
        *(float4*)&vs[r * 40 + c4] = z;
    }
    if (wave == 0) tensor_wait0();
#elif HAVE_ASYNC_LDS
    for (int i = tid; i < 49 * 8; i += 128) {
        const int r = i >> 3, c4 = (i & 7) * 4;
        const float* row = base + (size_t)r * 1536;
        async_copy16(row + c4,        &qs[r * 36 + c4]);
        async_copy16(row + 512 + c4,  &ks[r * 36 + c4]);
        async_copy16(row + 1024 + c4, &vs[r * 40 + c4]);
    }
    for (int i = tid; i < 15 * 8; i += 128) {
        const int r = 49 + (i >> 3), c4 = (i & 7) * 4;
        const float4 z = make_float4(0.f, 0.f, 0.f, 0.f);
        *(float4*)&qs[r * 36 + c4] = z;
        *(float4*)&ks[r * 36 + c4] = z;
        *(float4*)&vs[r * 40 + c4] = z;
    }
    async_wait0();
#else
    for (int i = tid; i < 64 * 8; i += 128) {
        const int r = i >> 3, c4 = (i & 7) * 4;
        float4 q4 = make_float4(0.f, 0.f, 0.f, 0.f), k4 = q4, v4 = q4;
        if (r < 49) {
            const float* row = base + (size_t)r * 1536;
            q4 = *(const float4*)(row + c4);
            k4 = *(const float4*)(row + 512 + c4);
            v4 = *(const float4*)(row + 1024 + c4);
        }
        *(float4*)&qs[r * 36 + c4] = q4;
        *(float4*)&ks[r * 36 + c4] = k4;
        *(float4*)&vs[r * 40 + c4] = v4;
    }
#endif
    __syncthreads();

    // ---- S = q @ k^T (64x64 padded, K = d = 32) ----
    v8f s[4] = {};
    #pragma unroll
    for (int kk = 0; kk < 32; kk += 4) {
        const int ko = kk + 2 * half;
        const float* ap = &qs[(m0 + l16) * 36 + ko];
        const v2f a = (v2f){ap[0], ap[1]};
        #pragma unroll
        for (int nt = 0; nt < 4; ++nt) {
            const float* bp = &ks[(nt * 16 + l16) * 36 + ko];   // B = k^T
            const v2f bb = (v2f){bp[0], bp[1]};
            s[nt] = WMMA_F32(a, bb, s[nt]);
        }
    }

    // ---- S*scale + relative position bias -> LDS ----
    #pragma unroll
    for (int nt = 0; nt < 4; ++nt) {
        #pragma unroll
        for (int j = 0; j < 8; ++j) {
            const int i  = m0 + j + 8 * half;
            const int jc = nt * 16 + l16;
            float val = s[nt][j] * scale;
            if (i < 49 && jc < 49) {
                const int ri = i / 7,  ci = i - ri * 7;
                const int rj = jc / 7, cj = jc - rj * 7;
                const int idx = (ri - rj + 6) * 13 + (ci - cj + 6);
                val += table[idx * 16 + h];
            }
            ps[i * 68 + jc] = val;
        }
    }
    __syncthreads();

    // ---- softmax over 49 keys (one thread per query row) ----
    if (tid < 49) {
        float* row = &ps[tid * 68];
        float mx = row[0];
        for (int j = 1; j < 49; ++j) mx = fmaxf(mx, row[j]);
        float sum = 0.f;
        for (int j = 0; j < 49; ++j) { float e = __expf(row[j] - mx); row[j] = e; sum += e; }
        const float inv = 1.0f / sum;
        for (int j = 0; j < 49; ++j) row[j] *= inv;
        for (int j = 49; j < 64; ++j) row[j] = 0.f;     // padded keys contribute 0
    } else if (tid < 64) {
        float* row = &ps[tid * 68];                     // padded query rows: all 0
        for (int j = 0; j < 64; ++j) row[j] = 0.f;
    }
    __syncthreads();

    // ---- O = P @ v  (rows x 32, K = 64 padded) ----
    v8f o[2] = {};
    #pragma unroll
    for (int kk = 0; kk < 64; kk += 4) {
        const int ko = kk + 2 * half;
        const float* ap = &ps[(m0 + l16) * 68 + ko];
        const v2f a = (v2f){ap[0], ap[1]};
        #pragma unroll
        for (int nt = 0; nt < 2; ++nt) {
            const int nc = nt * 16 + l16;
            const v2f bb = (v2f){vs[ko * 40 + nc], vs[(ko + 1) * 40 + nc]};
            o[nt] = WMMA_F32(a, bb, o[nt]);
        }
    }

    // ---- scatter O into (b, n, C) with col = h*32 + d ----
    #pragma unroll
    for (int nt = 0; nt < 2; ++nt) {
        #pragma unroll
        for (int j = 0; j < 8; ++j) {
            const int i = m0 + j + 8 * half;
            if (i < 49)
                obuf[((size_t)b * 49 + i) * 512 + h * 32 + nt * 16 + l16] = o[nt][j];
        }
    }
}

// ---------------------------------------------------------------------------
extern "C" void kernel_launch(void* const* d_in, const int* in_sizes, int n_in,
                              void* d_out, int out_size, void* d_ws, size_t ws_size,
                              hipStream_t stream) {
    const float* x      = (const float*)d_in[0];   // (nWB, 49, 512)
    const float* qkv_w  = (const float*)d_in[1];   // (512, 1536)
    const float* qkv_b  = (const float*)d_in[2];   // (1536)
    const float* table  = (const float*)d_in[3];   // (169, 16)
    const float* proj_w = (const float*)d_in[4];   // (512, 512)
    const float* proj_b = (const float*)d_in[5];   // (512)
    float* out = (float*)d_out;

    const int C    = 512;
    const int nWB  = in_sizes[0] / (49 * C);       // 1024
    const int Mrows = nWB * 49;                    // 50176 (multiple of 64)

    float* qkv  = (float*)d_ws;                    // (Mrows, 1536)
    float* obuf = qkv + (size_t)Mrows * 3 * C;     // (Mrows, 512)

    // 1) qkv = x @ qkv_w + qkv_b
    dim3 g1((3 * C) / 128, (Mrows + 63) / 64);
    gemm_wmma_f32<<<g1, 256, 0, stream>>>(x, qkv_w, qkv_b, qkv, Mrows, 3 * C, C);

    // 2) per-(window, head) attention -> obuf
    attn_wmma_f32<<<nWB * 16, 128, 0, stream>>>(qkv, table, obuf);

    // 3) out = obuf @ proj_w + proj_b
    dim3 g3(C / 128, (Mrows + 63) / 64);
    gemm_wmma_f32<<<g3, 256, 0, stream>>>(obuf, proj_w, proj_b, out, Mrows, C, C);
}